// CorrRatio_52862457480009
// MI455X (gfx1250) — compile-verified
//
#include <hip/hip_runtime.h>
#include <hip/hip_bf16.h>
#include <float.h>

// ---------------------------------------------------------------------------
// Symmetric correlation ratio for MI455X (gfx1250, wave32).
// Heavy Parzen accumulation is mapped onto V_WMMA_F32_16X16X4_F32:
//   D(16x16) += A(16 bins x 4 voxels) x B(4 voxels x 16 cols)
// col0 of B = 1.0 -> bin_counts, col1 = clipped x -> sum(w*x). Full fp32.
// Weights exp(-preterm*(y-vbc)^2) = exp2(-((y - vbc)*s)^2), s = sqrt(pt*log2e)
// so each weight is sub -> mul(neg) -> v_exp_f32. WMMA absorbs all MACs.
// ---------------------------------------------------------------------------

typedef __attribute__((ext_vector_type(2))) float v2f;
typedef __attribute__((ext_vector_type(8))) float v8f;

#define EPS_F32 1.1920929e-7f

#if defined(__has_builtin)
#if __has_builtin(__builtin_amdgcn_exp2f)
#define EXP2F(x) __builtin_amdgcn_exp2f(x)
#else
#define EXP2F(x) exp2f(x)
#endif
#else
#define EXP2F(x) exp2f(x)
#endif

__device__ __forceinline__ unsigned fkey(float f) {
    unsigned b = __float_as_uint(f);
    return (b & 0x80000000u) ? ~b : (b | 0x80000000u);
}
__device__ __forceinline__ float funkey(unsigned k) {
    unsigned b = (k & 0x80000000u) ? (k & 0x7FFFFFFFu) : ~k;
    return __uint_as_float(b);
}
__device__ __forceinline__ unsigned umaxu(unsigned a, unsigned b) { return a > b ? a : b; }

// ---------------------------------------------------------------------------
// Kernel 1: per-array min/max via ordered-uint keys. hdr[0]=~minkeyA,
// hdr[1]=maxkeyA, hdr[2]=~minkeyB, hdr[3]=maxkeyB. Zero-init is identity.
// ---------------------------------------------------------------------------
__global__ __launch_bounds__(256) void minmax_kernel(const float* __restrict__ a,
                                                     const float* __restrict__ b,
                                                     int n, unsigned* __restrict__ hdr) {
    unsigned mnA = 0, mxA = 0, mnB = 0, mxB = 0;
    int stride = gridDim.x * blockDim.x;
    for (int i = blockIdx.x * blockDim.x + threadIdx.x; i < n; i += stride) {
        unsigned ka = fkey(a[i]);
        mnA = umaxu(mnA, ~ka); mxA = umaxu(mxA, ka);
        unsigned kb = fkey(b[i]);
        mnB = umaxu(mnB, ~kb); mxB = umaxu(mxB, kb);
    }
    __shared__ unsigned sm[4][256];
    sm[0][threadIdx.x] = mnA; sm[1][threadIdx.x] = mxA;
    sm[2][threadIdx.x] = mnB; sm[3][threadIdx.x] = mxB;
    __syncthreads();
    for (int s = 128; s > 0; s >>= 1) {
        if ((int)threadIdx.x < s)
            for (int k = 0; k < 4; ++k)
                sm[k][threadIdx.x] = umaxu(sm[k][threadIdx.x], sm[k][threadIdx.x + s]);
        __syncthreads();
    }
    if (threadIdx.x == 0) {
        atomicMax(hdr + 0, sm[0][0]);
        atomicMax(hdr + 1, sm[1][0]);
        atomicMax(hdr + 2, sm[2][0]);
        atomicMax(hdr + 3, sm[3][0]);
    }
}

// ---------------------------------------------------------------------------
// Kernel 2: fine histograms of both arrays over [min,max].
// ---------------------------------------------------------------------------
__global__ __launch_bounds__(256) void hist_kernel(const float* __restrict__ a,
                                                   const float* __restrict__ b,
                                                   int n, const unsigned* __restrict__ hdr,
                                                   unsigned* __restrict__ histA,
                                                   unsigned* __restrict__ histB, int nb) {
    float loA = funkey(~hdr[0]), hiA = funkey(hdr[1]);
    float loB = funkey(~hdr[2]), hiB = funkey(hdr[3]);
    float invA = (float)nb / fmaxf(hiA - loA, 1e-30f);
    float invB = (float)nb / fmaxf(hiB - loB, 1e-30f);
    int stride = gridDim.x * blockDim.x;
    for (int i = blockIdx.x * blockDim.x + threadIdx.x; i < n; i += stride) {
        int ba = (int)((a[i] - loA) * invA);
        ba = ba < 0 ? 0 : (ba > nb - 1 ? nb - 1 : ba);
        atomicAdd(histA + ba, 1u);
        int bb = (int)((b[i] - loB) * invB);
        bb = bb < 0 ? 0 : (bb > nb - 1 ? nb - 1 : bb);
        atomicAdd(histB + bb, 1u);
    }
}

// ---------------------------------------------------------------------------
// Kernel 3: single block. Scan histograms, locate 1%/99% quantile ranks with
// in-bin linear interpolation, emit per-direction constants:
// cst[d*8 + {0..5}] = f_min, f_max, m_min, m_max, h, preterm
// ---------------------------------------------------------------------------
__global__ __launch_bounds__(1024) void quantile_kernel(const unsigned* __restrict__ histA,
                                                        const unsigned* __restrict__ histB,
                                                        int nb, const unsigned* __restrict__ hdr,
                                                        int n, float* __restrict__ cst) {
    __shared__ unsigned part[1024];
    __shared__ unsigned scan[1024];
    __shared__ float qv[4];  // qA01, qA99, qB01, qB99
    int t = threadIdx.x;
    int chunk = nb / 1024;
    for (int d = 0; d < 2; ++d) {
        const unsigned* hist = d ? histB : histA;
        float lo = funkey(~hdr[d * 2 + 0]);
        float hi = funkey(hdr[d * 2 + 1]);
        float width = fmaxf(hi - lo, 1e-30f) / (float)nb;
        unsigned s = 0;
        for (int j = 0; j < chunk; ++j) s += hist[t * chunk + j];
        part[t] = s;
        __syncthreads();
        if (t == 0) {
            unsigned c = 0;
            for (int i = 0; i < 1024; ++i) { scan[i] = c; c += part[i]; }
        }
        __syncthreads();
        for (int qi = 0; qi < 2; ++qi) {
            double pos = (qi ? 0.99 : 0.01) * (double)(n - 1);
            unsigned r = (unsigned)pos;
            if (part[t] != 0u && r >= scan[t] && r < scan[t] + part[t]) {
                unsigned c0 = scan[t];
                for (int j = 0; j < chunk; ++j) {
                    unsigned cnt = hist[t * chunk + j];
                    if (cnt != 0u && r < c0 + cnt) {
                        float frac = ((float)(pos - (double)c0) + 0.5f) / (float)cnt;
                        frac = fminf(frac, 1.0f);
                        qv[d * 2 + qi] = lo + ((float)(t * chunk + j) + frac) * width;
                        break;
                    }
                    c0 += cnt;
                }
            }
        }
        __syncthreads();
    }
    if (t == 0) {
        float qA01 = qv[0], qA99 = qv[1], qB01 = qv[2], qB99 = qv[3];
        for (int d = 0; d < 2; ++d) {
            // dir0: pred=A(y_true), target=B(y_pred); dir1 swapped.
            float f_min = d ? qA01 : qB01, f_max = d ? qA99 : qB99;
            float m_min = d ? qB01 : qA01, m_max = d ? qB99 : qA99;
            float h = (f_max - f_min) * (1.0f / 32.0f);
            float fs = h * (1.0f / 2.355f);
            float pre = 1.0f / (2.0f * fs * fs + 1e-38f);
            float* c = cst + d * 8;
            c[0] = f_min; c[1] = f_max; c[2] = m_min; c[3] = m_max;
            c[4] = h;     c[5] = pre;
        }
    }
}

// ---------------------------------------------------------------------------
// Kernel 4: Parzen accumulation via V_WMMA_F32_16X16X4_F32.
// A-matrix (16x4 fp32): lanes 0-15 hold K=0,1 in vgpr0/1, lanes 16-31 hold
// K=2,3; M = lane&15 = bin. B-matrix (4x16 fp32): vgpr0 = rows K0/K1,
// vgpr1 = rows K2/K3. acc_lo = bins 0-15, acc_hi = bins 16-31; D columns
// 0/1 carry sum_w / sum_wx. Main loop is branch/check free; tail voxels
// (n&3) handled by a single masked quad on global wave 0 (wave-uniform
// branch, EXEC all-ones around WMMA).
// ---------------------------------------------------------------------------
__global__ __launch_bounds__(256) void parzen_kernel(const float* __restrict__ pred,
                                                     const float* __restrict__ targ,
                                                     const float* __restrict__ cst,
                                                     float* __restrict__ partials,
                                                     int n) {
    const float f_min = cst[0], f_max = cst[1];
    const float m_min = cst[2], m_max = cst[3];
    const float h = cst[4], preterm = cst[5];

    const int lane   = threadIdx.x & 31;
    const int wave   = threadIdx.x >> 5;
    const int col    = lane & 15;     // bin (A) / output column (B)
    const int hiHalf = lane >> 4;     // K-pair selector

    // exp(-pt*d^2) = exp2(-(d*s)^2), s = sqrt(pt*log2(e))
    const float s = __builtin_sqrtf(preterm * 1.44269504088896340736f);
    const float vls = (f_min + h * ((float)col + 0.5f)) * s;   // vbc_lo * s
    const float vhs = vls + 16.0f * h * s;                     // vbc_hi * s

    const float m0 = (col == 0) ? 1.0f : 0.0f;   // ones column
    const float m1 = (col == 1) ? 1.0f : 0.0f;   // x column
    const float m2 = (col == 2) ? 1.0f : 0.0f;   // sum_x bookkeeping lanes

    const int gwave  = blockIdx.x * (blockDim.x >> 5) + wave;
    const int nwaves = gridDim.x * (blockDim.x >> 5);

    v8f acc_lo = {0.f, 0.f, 0.f, 0.f, 0.f, 0.f, 0.f, 0.f};
    v8f acc_hi = {0.f, 0.f, 0.f, 0.f, 0.f, 0.f, 0.f, 0.f};
    float sx = 0.f, sx2 = 0.f;

    // contiguous chunk of complete quads per wave
    const int nfull = n >> 2;
    const int per = (nfull + nwaves - 1) / nwaves;
    const int q0 = gwave * per;
    const int q1 = (q0 + per) < nfull ? (q0 + per) : nfull;

    for (int q = q0; q < q1; ++q) {
        const int v = q << 2;
        const float2 yp = *(const float2*)(targ + v + 2 * hiHalf);
        const float ya = fminf(fmaxf(yp.x, f_min), f_max);
        const float yb = fminf(fmaxf(yp.y, f_min), f_max);
        const float yas = ya * s, ybs = yb * s;
        const float d0l = yas - vls, d1l = ybs - vls;
        const float d0h = yas - vhs, d1h = ybs - vhs;
        v2f a_lo, a_hi;
        a_lo.x = EXP2F(-(d0l * d0l));
        a_lo.y = EXP2F(-(d1l * d1l));
        a_hi.x = EXP2F(-(d0h * d0h));
        a_hi.y = EXP2F(-(d1h * d1h));

        float x0 = pred[v + hiHalf];
        float x1 = pred[v + 2 + hiHalf];
        x0 = fminf(fmaxf(x0, m_min), m_max);
        x1 = fminf(fmaxf(x1, m_min), m_max);

        v2f b;
        b.x = fmaf(m1, x0, m0);
        b.y = fmaf(m1, x1, m0);

        acc_lo = __builtin_amdgcn_wmma_f32_16x16x4_f32(
            false, a_lo, false, b, (short)0, acc_lo, false, false);
        acc_hi = __builtin_amdgcn_wmma_f32_16x16x4_f32(
            false, a_hi, false, b, (short)0, acc_hi, false, false);

        sx  = fmaf(m2, x0 + x1, sx);
        sx2 = fmaf(m2, fmaf(x0, x0, x1 * x1), sx2);
    }

    // tail (n & 3) voxels: one masked quad on global wave 0 (uniform branch)
    if (gwave == 0 && (n & 3)) {
        const int v = nfull << 2;
        const int ia = v + 2 * hiHalf;
        const bool ok0 = ia < n, ok1 = (ia + 1) < n;
        float ya = targ[ok0 ? ia : n - 1];
        float yb = targ[ok1 ? ia + 1 : n - 1];
        ya = fminf(fmaxf(ya, f_min), f_max) * s;
        yb = fminf(fmaxf(yb, f_min), f_max) * s;
        const float d0l = ya - vls, d1l = yb - vls;
        const float d0h = ya - vhs, d1h = yb - vhs;
        v2f a_lo, a_hi;
        a_lo.x = ok0 ? EXP2F(-(d0l * d0l)) : 0.f;
        a_lo.y = ok1 ? EXP2F(-(d1l * d1l)) : 0.f;
        a_hi.x = ok0 ? EXP2F(-(d0h * d0h)) : 0.f;
        a_hi.y = ok1 ? EXP2F(-(d1h * d1h)) : 0.f;

        const int jb0 = v + hiHalf, jb1 = v + 2 + hiHalf;
        const bool okx0 = jb0 < n, okx1 = jb1 < n;
        float x0 = pred[okx0 ? jb0 : n - 1];
        float x1 = pred[okx1 ? jb1 : n - 1];
        x0 = fminf(fmaxf(x0, m_min), m_max);
        x1 = fminf(fmaxf(x1, m_min), m_max);

        v2f b;
        b.x = fmaf(m1, x0, m0);
        b.y = fmaf(m1, x1, m0);

        acc_lo = __builtin_amdgcn_wmma_f32_16x16x4_f32(
            false, a_lo, false, b, (short)0, acc_lo, false, false);
        acc_hi = __builtin_amdgcn_wmma_f32_16x16x4_f32(
            false, a_hi, false, b, (short)0, acc_hi, false, false);

        if (okx0) { sx += m2 * x0; sx2 += m2 * x0 * x0; }
        if (okx1) { sx += m2 * x1; sx2 += m2 * x1 * x1; }
    }

    // sum_x lives only in lanes 2 and 18: one xor-16 shuffle completes it
    sx  += __shfl_xor(sx, 16, 32);
    sx2 += __shfl_xor(sx2, 16, 32);

    // D layout: lanes 0-15 hold M=0..7 (vgpr j) for N=lane; lanes 16-31 M=8..15.
    // Columns 0/1 (lanes 0,16 / 1,17) carry sum_w / sum_wx per bin.
    __shared__ float red[8][66];
    if (col < 2) {
        float* dst = &red[wave][col * 32];
        const int mofs = hiHalf * 8;
        #pragma unroll
        for (int j = 0; j < 8; ++j) {
            dst[mofs + j]      = acc_lo[j];  // bins  0..15 (lo)
            dst[16 + mofs + j] = acc_hi[j];  // bins 16..31 (hi)
        }
    }
    if (lane == 2) { red[wave][64] = sx; red[wave][65] = sx2; }
    __syncthreads();
    if (threadIdx.x < 66) {
        float ssum = 0.f;
        #pragma unroll
        for (int w = 0; w < 8; ++w) ssum += red[w][threadIdx.x];
        partials[(size_t)blockIdx.x * 66 + threadIdx.x] = ssum;
    }
}

// ---------------------------------------------------------------------------
// Kernel 5: fixed-order final reduction -> eta^2 per direction -> output.
// ---------------------------------------------------------------------------
__global__ __launch_bounds__(256) void final_kernel(const float* __restrict__ part,
                                                    float* __restrict__ out,
                                                    int nblk, float Nf) {
    __shared__ float S[2][66];
    for (int idx = threadIdx.x; idx < 132; idx += blockDim.x) {
        int d = idx / 66, t = idx - d * 66;
        const float* p = part + (size_t)d * nblk * 66 + t;
        float s = 0.f;
        for (int b = 0; b < nblk; ++b) s += p[(size_t)b * 66];
        S[d][t] = s;
    }
    __syncthreads();
    if (threadIdx.x == 0) {
        float res = 0.f;
        for (int d = 0; d < 2; ++d) {
            float sx = S[d][64], sx2 = S[d][65];
            float tm = sx / Nf;
            float sumw = 0.f, bg = 0.f;
            for (int k = 0; k < 32; ++k) {
                float w = S[d][k], wx = S[d][32 + k];
                float mi = wx / (w + EPS_F32);
                float dd = mi - tm;
                bg = fmaf(w, dd * dd, bg);
                sumw += w;
            }
            float bgv = bg / (sumw + EPS_F32);
            float var = (sx2 - sx * sx / Nf) / (Nf - 1.0f);
            res += bgv / (var + EPS_F32);
        }
        out[0] = -0.5f * res;
    }
}

// ---------------------------------------------------------------------------
extern "C" void kernel_launch(void* const* d_in, const int* in_sizes, int n_in,
                              void* d_out, int out_size, void* d_ws, size_t ws_size,
                              hipStream_t stream) {
    const float* A = (const float*)d_in[0];  // y_true (pred in direction 0)
    const float* B = (const float*)d_in[1];  // y_pred (target in direction 0)
    float* out = (float*)d_out;
    const int n = in_sizes[0];

    // workspace layout (self-sizing to ws_size)
    const size_t HDR_OFF = 0;                // 4 uints
    const size_t CST_OFF = 256;              // 16 floats
    const size_t PART_OFF = 512;
    int NBLK = (ws_size >= (1u << 20)) ? 256 : 64;
    size_t partsz = (size_t)2 * NBLK * 66 * sizeof(float);
    size_t hist_off = (PART_OFF + partsz + 255) & ~(size_t)255;
    size_t nb = 1u << 20;
    while (nb > 4096 && hist_off + 2 * nb * sizeof(unsigned) > ws_size) nb >>= 1;
    size_t used = hist_off + 2 * nb * sizeof(unsigned);
    if (used > ws_size) used = ws_size;

    hipMemsetAsync(d_ws, 0, used, stream);

    char* ws = (char*)d_ws;
    unsigned* hdr = (unsigned*)(ws + HDR_OFF);
    float* cst = (float*)(ws + CST_OFF);
    float* part = (float*)(ws + PART_OFF);
    unsigned* histA = (unsigned*)(ws + hist_off);
    unsigned* histB = histA + nb;

    minmax_kernel<<<512, 256, 0, stream>>>(A, B, n, hdr);
    hist_kernel<<<512, 256, 0, stream>>>(A, B, n, hdr, histA, histB, (int)nb);
    quantile_kernel<<<1, 1024, 0, stream>>>(histA, histB, (int)nb, hdr, n, cst);

    // direction 0: pred = y_true, target = y_pred
    parzen_kernel<<<NBLK, 256, 0, stream>>>(A, B, cst + 0, part, n);
    // direction 1: pred = y_pred, target = y_true
    parzen_kernel<<<NBLK, 256, 0, stream>>>(B, A, cst + 8, part + (size_t)NBLK * 66, n);

    final_kernel<<<1, 256, 0, stream>>>(part, out, NBLK, (float)n);
}